// InformationLoss_5918464934402
// MI455X (gfx1250) — compile-verified
//
#include <hip/hip_runtime.h>
#include <hip/hip_bf16.h>
#include <math.h>

#define B_ROWS 1024
#define R_COLS 128

typedef __attribute__((ext_vector_type(2))) float v2f;
typedef __attribute__((ext_vector_type(8))) float v8f;

// Raw hardware exp2 (v_exp_f32) if available; library exp2f otherwise.
__device__ __forceinline__ float hw_exp2(float x) {
#if __has_builtin(__builtin_amdgcn_exp2f)
    return __builtin_amdgcn_exp2f(x);
#else
    return exp2f(x);
#endif
}

// ---------------------------------------------------------------------------
// Kernel 1: per-column mean and KDE bandwidth h = max(1.06*std*B^-0.2, 1e-4)
// 1024^(-0.2) == 0.25 exactly.
// ---------------------------------------------------------------------------
__global__ __launch_bounds__(256) void col_stats_kernel(
    const float* __restrict__ act,
    float* __restrict__ mean_out,
    float* __restrict__ h_out)
{
    __shared__ float ssum[256];
    __shared__ float ssq[256];
    const int r = blockIdx.x;
    const int t = threadIdx.x;

    float s = 0.0f, q = 0.0f;
    #pragma unroll
    for (int i = t; i < B_ROWS; i += 256) {
        float x = act[i * R_COLS + r];
        s += x;
        q += x * x;
    }
    ssum[t] = s; ssq[t] = q;
    __syncthreads();
    for (int off = 128; off >= 1; off >>= 1) {
        if (t < off) { ssum[t] += ssum[t + off]; ssq[t] += ssq[t + off]; }
        __syncthreads();
    }
    if (t == 0) {
        float mean = ssum[0] * (1.0f / (float)B_ROWS);
        float var  = (ssq[0] - (float)B_ROWS * mean * mean) * (1.0f / (float)(B_ROWS - 1));
        float sd   = sqrtf(fmaxf(var, 0.0f));
        float h    = fmaxf(1.06f * 0.25f * sd, 1.0e-4f);
        mean_out[r] = mean;
        h_out[r]    = h;
    }
}

// ---------------------------------------------------------------------------
// Kernel 2: reflected-KDE entropy partials.
// Grid (R_COLS, 4): blockIdx.x = column r, blockIdx.y = 256-row i-chunk.
// All three Gaussian args are expanded so each (i,j) term is FMA-form:
//   c2*(xi-xj)^2     = (p  + c2*xj^2) + (-q)*xj
//   c2*(xi+xj)^2     = (p  + c2*xj^2) + ( q)*xj
//   c2*(xi+xj-2)^2   = (p2 + c2*xj^2) + (q2)*xj
// with c2 = -log2(e)/(2h^2), p=c2*xi^2, q=2*c2*xi, p2=c2*(xi-2)^2,
// q2=2*c2*(xi-2); exponentials are raw hardware exp2.
// entsum_out[r*4+chunk] = sum over the chunk's i of log(density+1e-8).
// ---------------------------------------------------------------------------
__global__ __launch_bounds__(256) void kde_entropy_kernel(
    const float* __restrict__ act,
    const float* __restrict__ h_in,
    float* __restrict__ entsum_out)
{
    __shared__ float  xs[B_ROWS];
    __shared__ float2 xp[B_ROWS];   // {xj, c2*xj*xj}
    __shared__ float  red[256];
    const int r     = blockIdx.x;
    const int chunk = blockIdx.y;
    const int t     = threadIdx.x;

    // Stage the column into LDS (async global->LDS where the toolchain has it).
#if __has_builtin(__builtin_amdgcn_global_load_async_to_lds_b32)
    #pragma unroll
    for (int i = t; i < B_ROWS; i += 256) {
        __builtin_amdgcn_global_load_async_to_lds_b32(
            (__attribute__((address_space(1))) int*)(act + i * R_COLS + r),
            (__attribute__((address_space(3))) int*)&xs[i],
            0, 0);
    }
  #if __has_builtin(__builtin_amdgcn_s_wait_asynccnt)
    __builtin_amdgcn_s_wait_asynccnt(0);
  #else
    asm volatile("s_wait_asynccnt 0" ::: "memory");
  #endif
#else
    #pragma unroll
    for (int i = t; i < B_ROWS; i += 256) xs[i] = act[i * R_COLS + r];
#endif
    __syncthreads();

    const float h    = h_in[r];
    const float invh = 1.0f / h;
    const float c2   = -0.5f * invh * invh * 1.44269504088896340736f; // * log2(e)
    const float scale = invh * (1.0f / (float)B_ROWS) * 0.39894228040143267794f;

    #pragma unroll
    for (int i = t; i < B_ROWS; i += 256) {
        float v = xs[i];
        xp[i] = make_float2(v, c2 * v * v);
    }
    __syncthreads();

    // This thread's sample.
    const float xi  = xs[chunk * 256 + t];
    const float p   = c2 * xi * xi;
    const float q   = 2.0f * c2 * xi;
    const float qn  = -q;
    const float xm2 = xi - 2.0f;
    const float p2  = c2 * xm2 * xm2;
    const float q2  = 2.0f * c2 * xm2;

    float acc = 0.0f;
    #pragma unroll 8
    for (int j = 0; j < B_ROWS; ++j) {
        const float2 pj = xp[j];          // one broadcast ds_load_b64
        const float t1 = p  + pj.y;
        const float t2 = p2 + pj.y;
        acc += hw_exp2(fmaf(qn, pj.x, t1))
             + hw_exp2(fmaf(q,  pj.x, t1))
             + hw_exp2(fmaf(q2, pj.x, t2));
    }

    red[t] = __logf(fmaf(acc, scale, 1.0e-8f));
    __syncthreads();
    for (int off = 128; off >= 1; off >>= 1) {
        if (t < off) red[t] += red[t + off];
        __syncthreads();
    }
    if (t == 0) entsum_out[r * 4 + chunk] = red[0];
}

// ---------------------------------------------------------------------------
// Kernel 3: covariance off-diagonal squared sums via fp32 WMMA.
// cov = C^T C / (B-1), C = activity - mean (centered on the fly).
// One wave per 16x16 tile; fragment layout per ISA 7.12.2 (16x4 f32 A):
// lanes 0-15 hold K = k,k+1, lanes 16-31 hold K = k+2,k+3.
// ---------------------------------------------------------------------------
__global__ __launch_bounds__(32) void cov_tile_kernel(
    const float* __restrict__ act,
    const float* __restrict__ mean_in,
    float* __restrict__ tile_out)
{
    const int tile = blockIdx.x;
    const int tm = tile >> 3;
    const int tn = tile & 7;
    const int lane = threadIdx.x;     // wave32
    const int row  = lane & 15;
    const int hi   = lane >> 4;
    const int kb   = hi << 1;

    const int colA = tm * 16 + row;
    const int colB = tn * 16 + row;
    const float meanA = mean_in[colA];
    const float meanB = mean_in[colB];

    v8f c = {};

#if __has_builtin(__builtin_amdgcn_wmma_f32_16x16x4_f32)
    for (int k0 = 0; k0 < B_ROWS; k0 += 8) {
        v2f av0, bv0, av1, bv1;
        const int ka = (k0 + kb) * R_COLS;
        av0.x = act[ka          + colA] - meanA;
        av0.y = act[ka + R_COLS + colA] - meanA;
        bv0.x = act[ka          + colB] - meanB;
        bv0.y = act[ka + R_COLS + colB] - meanB;
        const int kc = (k0 + 4 + kb) * R_COLS;
        av1.x = act[kc          + colA] - meanA;
        av1.y = act[kc + R_COLS + colA] - meanA;
        bv1.x = act[kc          + colB] - meanB;
        bv1.y = act[kc + R_COLS + colB] - meanB;

        c = __builtin_amdgcn_wmma_f32_16x16x4_f32(false, av0, false, bv0,
                                                  (short)0, c, false, false);
        c = __builtin_amdgcn_wmma_f32_16x16x4_f32(false, av1, false, bv1,
                                                  (short)0, c, false, false);
    }
#else
    for (int v = 0; v < 8; ++v) {
        const int gm = tm * 16 + v + 8 * hi;
        const int gn = tn * 16 + row;
        const float mm = mean_in[gm], mn = mean_in[gn];
        float s = 0.0f;
        for (int k = 0; k < B_ROWS; ++k)
            s += (act[k * R_COLS + gm] - mm) * (act[k * R_COLS + gn] - mn);
        c[v] = s;
    }
#endif

    // Scale by 1/(B-1), square, reduce; diagonal masked only on diagonal tiles.
    const float inv = 1.0f / (float)(B_ROWS - 1);
    float part = 0.0f;
    if (tm == tn) {
        #pragma unroll
        for (int v = 0; v < 8; ++v) {
            float e = c[v] * inv;
            part += ((v + 8 * hi) == row) ? 0.0f : e * e;
        }
    } else {
        #pragma unroll
        for (int v = 0; v < 8; ++v) {
            float e = c[v] * inv;
            part += e * e;
        }
    }
    #pragma unroll
    for (int off = 16; off >= 1; off >>= 1)
        part += __shfl_xor(part, off);
    if (lane == 0) tile_out[tile] = part;
}

// ---------------------------------------------------------------------------
// Kernel 4: total = (1/(B*R)) * sum(entsum[0..511]) + sum(tiles[0..63])
// ---------------------------------------------------------------------------
__global__ __launch_bounds__(256) void finalize_kernel(
    const float* __restrict__ entsum,
    const float* __restrict__ tiles,
    float* __restrict__ out)
{
    __shared__ float red[256];
    const int t = threadIdx.x;
    float v = (entsum[t] + entsum[t + 256]) * (1.0f / ((float)B_ROWS * (float)R_COLS));
    if (t < 64) v += tiles[t];
    red[t] = v;
    __syncthreads();
    for (int off = 128; off >= 1; off >>= 1) {
        if (t < off) red[t] += red[t + off];
        __syncthreads();
    }
    if (t == 0) out[0] = red[0];
}

// ---------------------------------------------------------------------------
extern "C" void kernel_launch(void* const* d_in, const int* in_sizes, int n_in,
                              void* d_out, int out_size, void* d_ws, size_t ws_size,
                              hipStream_t stream)
{
    const float* act = (const float*)d_in[0];   // (1024, 128) f32
    float* ws     = (float*)d_ws;
    float* mean   = ws;          // 128 floats
    float* h      = ws + 128;    // 128 floats
    float* entsum = ws + 256;    // 512 floats (128 cols x 4 chunks)
    float* tiles  = ws + 768;    // 64 floats
    float* out    = (float*)d_out;

    col_stats_kernel  <<<R_COLS, 256, 0, stream>>>(act, mean, h);
    kde_entropy_kernel<<<dim3(R_COLS, 4), 256, 0, stream>>>(act, h, entsum);
    cov_tile_kernel   <<<64, 32, 0, stream>>>(act, mean, tiles);
    finalize_kernel   <<<1, 256, 0, stream>>>(entsum, tiles, out);
}